// sAMPpredGATEncoder_63608465654523
// MI455X (gfx1250) — compile-verified
//
#include <hip/hip_runtime.h>
#include <hip/hip_bf16.h>

// Problem constants (from reference)
#define HH     4
#define DHH    64
#define DD     256
#define NLAYER 3
#define BB     8
#define LL     1024
#define MROWS  (BB*LL)   // 8192

typedef __attribute__((ext_vector_type(2))) float v2f;
typedef __attribute__((ext_vector_type(8))) float v8f;

// ---------------------------------------------------------------------------
// Embedding: x[b,n,:] = embed[tok[b,n],:] + pos[n,:]
// ---------------------------------------------------------------------------
__global__ void embed_kernel(const int* __restrict__ tokens,
                             const float* __restrict__ embed,
                             const float* __restrict__ pos,
                             float* __restrict__ x) {
    int row = blockIdx.x;          // b*L + n
    int t   = threadIdx.x;         // 0..255
    int n   = row & (LL - 1);
    int tok = tokens[row];
    x[(size_t)row * DD + t] = embed[(size_t)tok * DD + t] + pos[(size_t)n * DD + t];
}

// ---------------------------------------------------------------------------
// LDS-tiled transpose: out[n*K + k] = in[k*N + n]   (per z-slice of K*N)
// block (32,8), grid (N/32, K/32, Z)
// ---------------------------------------------------------------------------
__global__ void transpose_kernel(const float* __restrict__ in,
                                 float* __restrict__ out, int K, int N) {
    __shared__ float tile[32][33];
    in  += (size_t)blockIdx.z * K * N;
    out += (size_t)blockIdx.z * K * N;
    int n0 = blockIdx.x * 32;
    int k0 = blockIdx.y * 32;
    int tx = threadIdx.x, ty = threadIdx.y;
    #pragma unroll
    for (int i = ty; i < 32; i += 8)
        tile[i][tx] = in[(size_t)(k0 + i) * N + (n0 + tx)];
    __syncthreads();
    #pragma unroll
    for (int i = ty; i < 32; i += 8)
        out[(size_t)(n0 + i) * K + (k0 + tx)] = tile[tx][i];
}

// ---------------------------------------------------------------------------
// fp32 WMMA GEMM: C[M,N] = act(A[M,K] @ BT^T + bias), BT is N-major [N,K].
// One wave computes a 64x16 strip (4 M-tiles); A and B fragments are b64 loads.
// ACT 0: none, 1: ELU
// ---------------------------------------------------------------------------
template<int ACT>
__global__ void gemm_kernel(const float* __restrict__ A,
                            const float* __restrict__ BT,
                            const float* __restrict__ bias,
                            float* __restrict__ C,
                            int M, int N, int K) {
    int wid  = (blockIdx.x * blockDim.x + threadIdx.x) >> 5;
    int lane = threadIdx.x & 31;
    int tiles_n = N >> 4;
    int tiles_m = M >> 6;
    if (wid >= tiles_m * tiles_n) return;
    int tm = wid / tiles_n;
    int tn = wid - tm * tiles_n;

    int li = lane & 15;
    int kg = lane >> 4;
    int m0 = tm * 64 + li;
    int n  = tn * 16 + li;
    const float* btp = BT + (size_t)n * K;

    v8f acc[4] = {v8f{}, v8f{}, v8f{}, v8f{}};

    for (int kk = 0; kk < K; kk += 4) {
        int ka = kk + 2 * kg;
        const float2 bw = *(const float2*)(btp + ka);
        v2f bv; bv.x = bw.x; bv.y = bw.y;
        if (kk + 32 < K) __builtin_prefetch(btp + ka + 32, 0, 1);
        #pragma unroll
        for (int t = 0; t < 4; t++) {
            const float2 av = *(const float2*)(A + (size_t)(m0 + t * 16) * K + ka);
            v2f af; af.x = av.x; af.y = av.y;
            acc[t] = __builtin_amdgcn_wmma_f32_16x16x4_f32(
                false, af, false, bv, (short)0, acc[t], false, false);
        }
    }

    #pragma unroll
    for (int t = 0; t < 4; t++) {
        int mb = tm * 64 + t * 16 + 8 * kg;
        #pragma unroll
        for (int r = 0; r < 8; r++) {
            float v = acc[t][r];
            if (bias) v += bias[n];
            if (ACT == 1) v = (v > 0.f) ? v : expm1f(v);
            C[(size_t)(mb + r) * N + n] = v;
        }
    }
}

// ---------------------------------------------------------------------------
// Projection GEMM with TRANSPOSED output:
//   hT[(b*H + head)*DH + e][j] = (x @ Wg)[b*L+j][head*64+e]
// D-fragment rows are consecutive j's per lane -> two aligned b128 stores/tile.
// ---------------------------------------------------------------------------
__global__ void proj_kernel(const float* __restrict__ A,    // x [M, 256]
                            const float* __restrict__ BT,   // WgT [256, 256] N-major
                            float* __restrict__ hT) {       // [2048, 1024]
    int wid  = (blockIdx.x * blockDim.x + threadIdx.x) >> 5; // 2048 waves
    int lane = threadIdx.x & 31;
    int tm = wid >> 4;            // 0..127 (64-row strips)
    int tn = wid & 15;            // 0..15

    int li = lane & 15;
    int kg = lane >> 4;
    int m0 = tm * 64 + li;
    int n  = tn * 16 + li;        // concat col: head = n>>6, e = n&63
    const float* btp = BT + (size_t)n * DD;

    v8f acc[4] = {v8f{}, v8f{}, v8f{}, v8f{}};

    for (int kk = 0; kk < DD; kk += 4) {
        int ka = kk + 2 * kg;
        const float2 bw = *(const float2*)(btp + ka);
        v2f bv; bv.x = bw.x; bv.y = bw.y;
        #pragma unroll
        for (int t = 0; t < 4; t++) {
            const float2 av = *(const float2*)(A + (size_t)(m0 + t * 16) * DD + ka);
            v2f af; af.x = av.x; af.y = av.y;
            acc[t] = __builtin_amdgcn_wmma_f32_16x16x4_f32(
                false, af, false, bv, (short)0, acc[t], false, false);
        }
    }

    int b    = (tm * 64) >> 10;   // 64-row strip never crosses a batch boundary
    int head = n >> 6, e = n & 63;
    float* colbase = hT + ((size_t)((b * HH + head) * DHH + e)) * LL;
    #pragma unroll
    for (int t = 0; t < 4; t++) {
        int nn = (tm * 64 + t * 16 + 8 * kg) & (LL - 1);   // first of 8 consecutive j's
        float4 lo, hi;
        lo.x = acc[t][0]; lo.y = acc[t][1]; lo.z = acc[t][2]; lo.w = acc[t][3];
        hi.x = acc[t][4]; hi.y = acc[t][5]; hi.z = acc[t][6]; hi.w = acc[t][7];
        *(float4*)(colbase + nn)     = lo;
        *(float4*)(colbase + nn + 4) = hi;
    }
}

// ---------------------------------------------------------------------------
// s[bh,n] = h_col-dot with a_src, d likewise; reads hT (coalesced across lanes)
// ---------------------------------------------------------------------------
__global__ void sd_kernel(const float* __restrict__ hT,
                          const float* __restrict__ asrc,
                          const float* __restrict__ adst,
                          float* __restrict__ sbuf,
                          float* __restrict__ dbuf) {
    int idx = blockIdx.x * blockDim.x + threadIdx.x;   // (b*H + h)*L + n
    int n  = idx & (LL - 1);
    int bh = idx >> 10;
    int hh = bh & (HH - 1);
    const float* hp = hT + (size_t)bh * DHH * LL + n;  // element e at hp[e*LL]
    const float* as = asrc + hh * DHH;
    const float* ad = adst + hh * DHH;
    float ss = 0.f, dd = 0.f;
    #pragma unroll 8
    for (int e = 0; e < DHH; e++) {
        float hv = hp[(size_t)e * LL];
        ss += hv * as[e];
        dd += hv * ad[e];
    }
    sbuf[idx] = ss;
    dbuf[idx] = dd;
}

// ---------------------------------------------------------------------------
// Softmax stats per attention row i (branchless two-pass):
//   m_i = max_j lrelu(s_i + d_j) over valid j,  Z_i = sum exp(.-m_i)
// lrelu(x) = max(x, 0.2x); masked logits -> -1e30 so exp underflows to 0.
// ---------------------------------------------------------------------------
__global__ void softmax_stats_kernel(const float* __restrict__ sbuf,
                                     const float* __restrict__ dbuf,
                                     const int*   __restrict__ mask,
                                     float* __restrict__ mrow,
                                     float* __restrict__ zrow) {
    int idx = blockIdx.x * blockDim.x + threadIdx.x;   // (b*H + h)*L + i
    int bh  = idx >> 10;
    int b   = bh >> 2;
    float si = sbuf[idx];
    const float2* dp = (const float2*)(dbuf + (size_t)bh * LL);
    const int2*   mk = (const int2*)(mask + (size_t)b * LL);

    float m = -1e30f;
    for (int j2 = 0; j2 < LL / 2; j2++) {
        float2 dv = dp[j2];
        int2   mv = mk[j2];
        float t0 = si + dv.x, t1 = si + dv.y;
        float e0 = fmaxf(t0, 0.2f * t0);
        float e1 = fmaxf(t1, 0.2f * t1);
        e0 = mv.x ? e0 : -1e30f;
        e1 = mv.y ? e1 : -1e30f;
        m = fmaxf(m, fmaxf(e0, e1));
    }
    float Z = 0.f;
    for (int j2 = 0; j2 < LL / 2; j2++) {
        float2 dv = dp[j2];
        int2   mv = mk[j2];
        float t0 = si + dv.x, t1 = si + dv.y;
        float e0 = fmaxf(t0, 0.2f * t0);
        float e1 = fmaxf(t1, 0.2f * t1);
        e0 = mv.x ? e0 : -1e30f;
        e1 = mv.y ? e1 : -1e30f;
        Z += __expf(e0 - m) + __expf(e1 - m);
    }
    mrow[idx] = m;
    zrow[idx] = 1.0f / Z;
}

// ---------------------------------------------------------------------------
// att = alpha @ h via WMMA; alpha regenerated branchlessly.
// One wave per (b,h, 16-row i-tile) -> full 16x64 strip. B fragments are
// contiguous b64 loads from hT; each alpha fragment feeds 4 v_wmma ops.
// ---------------------------------------------------------------------------
__global__ void attn_av_kernel(const float* __restrict__ hT,
                               const float* __restrict__ sbuf,
                               const float* __restrict__ dbuf,
                               const float* __restrict__ mrow,
                               const float* __restrict__ zrow,
                               const int*   __restrict__ mask,
                               float* __restrict__ attb) {
    int wid  = (blockIdx.x * blockDim.x + threadIdx.x) >> 5;  // 2048 waves
    int lane = threadIdx.x & 31;
    int it = wid & 63;           // i-tile (L/16 = 64)
    int bh = wid >> 6;           // 0..31
    int b  = bh >> 2;
    int hh = bh & 3;

    int li = lane & 15;
    int kg = lane >> 4;
    int i  = it * 16 + li;
    int ridx = bh * LL + i;
    float si = sbuf[ridx];
    float mi = mrow[ridx];
    float zi = zrow[ridx];
    const float2* dp  = (const float2*)(dbuf + (size_t)bh * LL);
    const int2*   mk  = (const int2*)(mask + (size_t)b * LL);
    const float*  hTb = hT + (size_t)bh * DHH * LL + (size_t)li * LL;  // col e=li base

    v8f acc[4] = {v8f{}, v8f{}, v8f{}, v8f{}};

    for (int jj = 0; jj < LL; jj += 4) {
        int ja = jj + 2 * kg;
        float2 dv = dp[ja >> 1];
        int2   mv = mk[ja >> 1];
        float t0 = si + dv.x;
        float e0 = fmaxf(t0, 0.2f * t0);
        float p0 = __expf(e0 - mi) * zi;
        float t1 = si + dv.y;
        float e1 = fmaxf(t1, 0.2f * t1);
        float p1 = __expf(e1 - mi) * zi;
        v2f a;
        a.x = mv.x ? p0 : 0.0f;
        a.y = mv.y ? p1 : 0.0f;

        #pragma unroll
        for (int et = 0; et < 4; et++) {
            const float2 hv = *(const float2*)(hTb + (size_t)(et * 16) * LL + ja);
            v2f bv; bv.x = hv.x; bv.y = hv.y;
            acc[et] = __builtin_amdgcn_wmma_f32_16x16x4_f32(
                false, a, false, bv, (short)0, acc[et], false, false);
        }
    }

    int mb = it * 16 + 8 * kg;
    #pragma unroll
    for (int et = 0; et < 4; et++) {
        float* op = attb + ((size_t)(b * LL + mb)) * DD + hh * DHH + et * 16 + li;
        #pragma unroll
        for (int r = 0; r < 8; r++) op[(size_t)r * DD] = acc[et][r];
    }
}

// ---------------------------------------------------------------------------
// out = LayerNorm(x + r) * gamma + beta   (one 256-thread block per row)
// ---------------------------------------------------------------------------
__global__ void add_ln_kernel(const float* __restrict__ xin,
                              const float* __restrict__ res,
                              const float* __restrict__ gam,
                              const float* __restrict__ bet,
                              float* __restrict__ out) {
    int row = blockIdx.x;
    int t   = threadIdx.x;
    float v = xin[(size_t)row * DD + t] + res[(size_t)row * DD + t];

    float sum = v, sq = v * v;
    #pragma unroll
    for (int o = 16; o; o >>= 1) {
        sum += __shfl_xor(sum, o, 32);
        sq  += __shfl_xor(sq,  o, 32);
    }
    __shared__ float wsum[8], wsq[8];
    int wv = t >> 5, ln = t & 31;
    if (ln == 0) { wsum[wv] = sum; wsq[wv] = sq; }
    __syncthreads();
    float tot = 0.f, totq = 0.f;
    #pragma unroll
    for (int i = 0; i < 8; i++) { tot += wsum[i]; totq += wsq[i]; }
    float mean = tot * (1.0f / DD);
    float var  = totq * (1.0f / DD) - mean * mean;
    float inv  = rsqrtf(var + 1e-5f);
    out[(size_t)row * DD + t] = (v - mean) * inv * gam[t] + bet[t];
}

// ---------------------------------------------------------------------------
extern "C" void kernel_launch(void* const* d_in, const int* in_sizes, int n_in,
                              void* d_out, int out_size, void* d_ws, size_t ws_size,
                              hipStream_t stream) {
    (void)in_sizes; (void)n_in; (void)out_size; (void)ws_size;
    const int*   tokens = (const int*)  d_in[0];
    const int*   mask   = (const int*)  d_in[1];
    const float* embedw = (const float*)d_in[2];
    const float* pos    = (const float*)d_in[3];
    const float* Wg     = (const float*)d_in[4];
    const float* asrc   = (const float*)d_in[5];
    const float* adst   = (const float*)d_in[6];
    const float* ln1s   = (const float*)d_in[7];
    const float* ln1b   = (const float*)d_in[8];
    const float* w1     = (const float*)d_in[9];
    const float* b1     = (const float*)d_in[10];
    const float* w2     = (const float*)d_in[11];
    const float* b2     = (const float*)d_in[12];
    const float* ln2s   = (const float*)d_in[13];
    const float* ln2b   = (const float*)d_in[14];

    // Workspace layout (floats): ~52 MB total
    float* x    = (float*)d_ws;                       // [M, D]
    float* hT   = x    + (size_t)MROWS * DD;          // [B*H*DH, L] = [2048, 1024]
    float* attb = hT   + (size_t)MROWS * DD;          // [M, D]
    float* ffh  = attb + (size_t)MROWS * DD;          // [M, 2D]
    float* ffo  = ffh  + (size_t)MROWS * 2 * DD;      // [M, D]
    float* sb   = ffo  + (size_t)MROWS * DD;          // [B*H*L]
    float* db   = sb   + (size_t)BB * HH * LL;
    float* mr   = db   + (size_t)BB * HH * LL;
    float* zr   = mr   + (size_t)BB * HH * LL;
    float* wgT  = zr   + (size_t)BB * HH * LL;        // [256, 256]
    float* w1T  = wgT  + (size_t)DD * DD;             // [512, 256]
    float* w2T  = w1T  + (size_t)DD * 2 * DD;         // [256, 512]

    embed_kernel<<<MROWS, DD, 0, stream>>>(tokens, embedw, pos, x);

    for (int l = 0; l < NLAYER; l++) {
        const float* Wgl = Wg + (size_t)l * HH * DD * DHH;
        // Per-head transpose of Wg: wgT[(head*64+e)*256 + k] = Wg[head][k][e]
        transpose_kernel<<<dim3(DHH / 32, DD / 32, HH), dim3(32, 8), 0, stream>>>(
            Wgl, wgT, DD, DHH);
        proj_kernel<<<256, 256, 0, stream>>>(x, wgT, hT);
        sd_kernel<<<(BB * HH * LL) / 256, 256, 0, stream>>>(
            hT, asrc + l * HH * DHH, adst + l * HH * DHH, sb, db);
        softmax_stats_kernel<<<(BB * HH * LL) / 256, 256, 0, stream>>>(sb, db, mask, mr, zr);
        attn_av_kernel<<<256, 256, 0, stream>>>(hT, sb, db, mr, zr, mask, attb);
        add_ln_kernel<<<MROWS, DD, 0, stream>>>(x, attb, ln1s + l * DD, ln1b + l * DD, x);
        // FF (weights pre-transposed to N-major)
        transpose_kernel<<<dim3(2 * DD / 32, DD / 32, 1), dim3(32, 8), 0, stream>>>(
            w1 + (size_t)l * DD * 2 * DD, w1T, DD, 2 * DD);
        gemm_kernel<1><<<512, 256, 0, stream>>>(
            x, w1T, b1 + (size_t)l * 2 * DD, ffh, MROWS, 2 * DD, DD);
        transpose_kernel<<<dim3(DD / 32, 2 * DD / 32, 1), dim3(32, 8), 0, stream>>>(
            w2 + (size_t)l * 2 * DD * DD, w2T, 2 * DD, DD);
        gemm_kernel<0><<<256, 256, 0, stream>>>(
            ffh, w2T, b2 + (size_t)l * DD, ffo, MROWS, DD, 2 * DD);
        add_ln_kernel<<<MROWS, DD, 0, stream>>>(x, ffo, ln2s + l * DD, ln2b + l * DD, x);
    }

    hipMemcpyAsync(d_out, x, (size_t)MROWS * DD * sizeof(float),
                   hipMemcpyDeviceToDevice, stream);
}